// InceptionPointTransformer_7378753815019
// MI455X (gfx1250) — compile-verified
//
#include <hip/hip_runtime.h>
#include <hip/hip_bf16.h>

typedef __attribute__((ext_vector_type(2))) float v2f;
typedef __attribute__((ext_vector_type(8))) float v8f;

#define NPTS   50000
#define CCH    64
#define NBR    3          // dilation branches
#define KMAX   64
#define KNN    16
#define NCLOUD 25
#define PTS    (NPTS / NCLOUD)   // 2000

// ---------------------------------------------------------------------------
// Kernel 1: fused projection GEMM  Y[g][N][64] = X @ W_g  for g = branch*3+{q,k,v}
// FP32 WMMA 16x16x4 keeps exact fp32 numerics (workload is bandwidth bound).
// Block: 288 threads = 9 waves; block handles 16 rows x all 576 output columns.
// ---------------------------------------------------------------------------
__global__ __launch_bounds__(288) void qkv_gemm(
    const float* __restrict__ x,
    const float* __restrict__ Wq, const float* __restrict__ Wk,
    const float* __restrict__ Wv,
    float* __restrict__ y)   // [9][NPTS][64]
{
    __shared__ __align__(16) float ldsA[16 * 68];   // 16 rows x 64, pad to 68

    const int t    = threadIdx.x;
    const int row0 = blockIdx.x * 16;

    // Stage A tile (16x64 contiguous floats) into LDS via float4 loads.
    if (t < 256) {
        const float4 v = ((const float4*)(x + (size_t)row0 * CCH))[t];
        const int e = t * 4;
        const int m = e >> 6;
        const int c = e & 63;
        float* p = &ldsA[m * 68 + c];
        p[0] = v.x; p[1] = v.y; p[2] = v.z; p[3] = v.w;
    }
    __syncthreads();

    const int lane  = t & 31;
    const int wave  = t >> 5;        // 0..8
    const int m     = lane & 15;     // A row within tile
    const int khalf = lane >> 4;     // 0/1 -> K lane-half split
    const int n     = lane & 15;     // B/D column within tile

    // Preload this lane's A fragments for all 16 K-steps (reused by 4 col tiles).
    v2f afrag[16];
#pragma unroll
    for (int kk = 0; kk < 16; ++kk) {
        const int kb = 4 * kk + 2 * khalf;
        afrag[kk] = *(const v2f*)&ldsA[m * 68 + kb];
    }

#pragma unroll
    for (int r = 0; r < 4; ++r) {
        const int ct     = wave + 9 * r;   // 0..35 column tile
        const int n0     = ct * 16;
        const int g      = n0 >> 6;        // group 0..8 = branch*3 + which
        const int coff   = n0 & 63;
        const int branch = g / 3;
        const int which  = g - 3 * branch;
        const float* Wg  = (which == 0 ? Wq : (which == 1 ? Wk : Wv))
                           + (size_t)branch * CCH * CCH;

        v8f acc = {};
#pragma unroll
        for (int kk = 0; kk < 16; ++kk) {
            const int kb = 4 * kk + 2 * khalf;
            v2f b;
            b.x = Wg[(size_t)(kb + 0) * CCH + coff + n];
            b.y = Wg[(size_t)(kb + 1) * CCH + coff + n];
            acc = __builtin_amdgcn_wmma_f32_16x16x4_f32(
                false, afrag[kk], false, b, (short)0, acc, false, false);
        }

        // D layout: VGPR rv -> row (khalf*8 + rv), col = n.
        float* outg = y + (size_t)g * NPTS * CCH;
#pragma unroll
        for (int rv = 0; rv < 8; ++rv) {
            const int mrow = khalf * 8 + rv;
            outg[(size_t)(row0 + mrow) * CCH + coff + n] = acc[rv];
        }
    }
}

// ---------------------------------------------------------------------------
// Kernel 2: per-cloud global max pool over x  ->  pool[25][64]
// ---------------------------------------------------------------------------
__global__ __launch_bounds__(256) void pool_kernel(
    const float* __restrict__ x, float* __restrict__ pool)
{
    __shared__ float red[256];
    const int cloud = blockIdx.x;
    const int c     = threadIdx.x & 63;
    const int chunk = threadIdx.x >> 6;      // 0..3
    const int span  = PTS / 4;               // 500

    const float* base = x + (size_t)cloud * PTS * CCH;
    float m = -INFINITY;
    for (int p = chunk * span; p < (chunk + 1) * span; ++p)
        m = fmaxf(m, base[(size_t)p * CCH + c]);
    red[threadIdx.x] = m;
    __syncthreads();
    if (threadIdx.x < 64) {
        m = fmaxf(fmaxf(red[c], red[64 + c]),
                  fmaxf(red[128 + c], red[192 + c]));
        pool[cloud * CCH + c] = m;
    }
}

// ---------------------------------------------------------------------------
// Kernel 3: fused point-transformer attention, branch max, pool max, residual.
// 64 threads (= channels) per node, 4 nodes per 256-thread block.
// Online softmax over 16 dilated neighbors per branch.
// ---------------------------------------------------------------------------
__global__ __launch_bounds__(256) void attn_kernel(
    const float* __restrict__ x,   const float* __restrict__ pos,
    const float* __restrict__ Wp,  const float* __restrict__ bp,
    const int*   __restrict__ esrc,const int*   __restrict__ batch,
    const float* __restrict__ qkv, const float* __restrict__ pool,
    float* __restrict__ out)
{
    __shared__ int ldsJ[4 * KMAX];
    const int ln = threadIdx.x >> 6;   // local node 0..3
    const int c  = threadIdx.x & 63;   // channel
    const int i  = blockIdx.x * 4 + ln;

    // Stage this node's 64 neighbor indices into LDS (one load per thread).
    ldsJ[ln * KMAX + c] = esrc[(size_t)i * KMAX + c];
    __syncthreads();

    const float xc  = x[(size_t)i * CCH + c];
    const float pix = pos[i * 3 + 0];
    const float piy = pos[i * 3 + 1];
    const float piz = pos[i * 3 + 2];

    float res = -INFINITY;
#pragma unroll
    for (int b = 0; b < NBR; ++b) {
        const int d = 1 << b;   // dilation 1,2,4
        const float* Q  = qkv + (size_t)(3 * b + 0) * NPTS * CCH;
        const float* Kk = qkv + (size_t)(3 * b + 1) * NPTS * CCH;
        const float* V  = qkv + (size_t)(3 * b + 2) * NPTS * CCH;

        const float qc  = Q[(size_t)i * CCH + c];
        const float wp0 = Wp[b * 192 + 0 * CCH + c];
        const float wp1 = Wp[b * 192 + 1 * CCH + c];
        const float wp2 = Wp[b * 192 + 2 * CCH + c];
        const float bpc = bp[b * CCH + c];

        float m = -INFINITY, s = 0.f, acc = 0.f;
#pragma unroll 4
        for (int tt = 0; tt < KNN; ++tt) {
            const int j = ldsJ[ln * KMAX + tt * d];
            if (tt + 1 < KNN) {   // hide gather latency: prefetch next k/v rows
                const int jn = ldsJ[ln * KMAX + (tt + 1) * d];
                __builtin_prefetch(&Kk[(size_t)jn * CCH + c], 0, 3);
                __builtin_prefetch(&V [(size_t)jn * CCH + c], 0, 3);
            }
            const float dx = pix - pos[j * 3 + 0];
            const float dy = piy - pos[j * 3 + 1];
            const float dz = piz - pos[j * 3 + 2];
            const float delta = fmaf(dx, wp0, fmaf(dy, wp1, fmaf(dz, wp2, bpc)));
            const float l  = qc - Kk[(size_t)j * CCH + c] + delta;
            const float nm = fmaxf(m, l);
            const float sc = __expf(m - nm);   // exp(-inf)=0 on first iter
            const float w  = __expf(l - nm);
            s   = fmaf(s,   sc, w);
            acc = fmaf(acc, sc, w * (V[(size_t)j * CCH + c] + delta));
            m = nm;
        }
        res = fmaxf(res, acc / (s + 1e-16f));
    }

    const float pv = pool[batch[i] * CCH + c];
    out[(size_t)i * CCH + c] = fmaxf(res, pv) + xc;
}

// ---------------------------------------------------------------------------
extern "C" void kernel_launch(void* const* d_in, const int* in_sizes, int n_in,
                              void* d_out, int out_size, void* d_ws, size_t ws_size,
                              hipStream_t stream) {
    const float* x     = (const float*)d_in[0];
    const float* pos   = (const float*)d_in[1];
    const float* Wv    = (const float*)d_in[2];
    const float* Wq    = (const float*)d_in[3];
    const float* Wk    = (const float*)d_in[4];
    const float* Wp    = (const float*)d_in[5];
    const float* bp    = (const float*)d_in[6];
    const int*   esrc  = (const int*)d_in[7];          // row 0 = src
    const int*   batch = (const int*)d_in[8];
    float*       out   = (float*)d_out;

    float* qkv  = (float*)d_ws;                        // 9 * N * 64 floats
    float* pool = qkv + (size_t)9 * NPTS * CCH;        // + 25 * 64 floats

    qkv_gemm <<<NPTS / 16, 288, 0, stream>>>(x, Wq, Wk, Wv, qkv);
    pool_kernel<<<NCLOUD,   256, 0, stream>>>(x, pool);
    attn_kernel<<<NPTS / 4, 256, 0, stream>>>(x, pos, Wp, bp, esrc, batch,
                                              qkv, pool, out);
}